// PostProcessor_17154099380466
// MI455X (gfx1250) — compile-verified
//
#include <hip/hip_runtime.h>
#include <hip/hip_bf16.h>
#include <math.h>

// ---------------------------------------------------------------------------
// Mask R-CNN PostProcessor for MI455X (gfx1250, wave32)
//   K1: row softmax, denominator reduced via V_WMMA_F32_16X16X4_F32 (ones-B),
//       probabilities written transposed for coalesced per-class gathers
//   K2: per-class decode + bitonic sort + greedy NMS over survivors (LDS);
//       proposals staged global->LDS by the Tensor Data Mover, overlapped
//       with the sort phase (TENSORcnt)
//   K3: global top-100 selection + gather
// ---------------------------------------------------------------------------

#define NN 1024
#define CC 81
#define WIMG 1333.0f
#define HIMG 800.0f
#define SCORE_TH 0.05f
#define NMS_TH 0.5f
#define DETS 100
// log(1000/16)
#define XFORM_CLIP 4.135166556742356f

typedef __attribute__((ext_vector_type(2))) float        v2f;
typedef __attribute__((ext_vector_type(8))) float        v8f;
typedef __attribute__((ext_vector_type(4))) unsigned int v4u;
typedef __attribute__((ext_vector_type(4))) int          v4i;
typedef __attribute__((ext_vector_type(8))) int          v8i;

#if __has_builtin(__builtin_amdgcn_tensor_load_to_lds) && \
    __has_builtin(__builtin_amdgcn_s_wait_tensorcnt)
#define HAVE_TDM 1
#else
#define HAVE_TDM 0
#endif

// ---------------------------------------------------------------------------
// Kernel 1: softmax over rows of [NN, CC].
// One wave (32 lanes) per 16-row tile. Row-sum of exp() runs on the matrix
// pipe: D = A(16x4 exp chunk) * B(4x16 ones) accumulated over 21 k-chunks
// (81 padded to 84) in full f32, so the prob > 0.05 threshold decisions
// match the reference. Output is written transposed: probT[c*NN + row].
// ---------------------------------------------------------------------------
__global__ __launch_bounds__(32) void softmax_rows(const float* __restrict__ logits,
                                                   float* __restrict__ probT) {
    __shared__ float s_exp[16][84];
    __shared__ float s_sum[16];

    const int lane = threadIdx.x;          // 0..31
    const int r    = lane & 15;            // row within tile
    const int half = lane >> 4;            // which K-pair this lane feeds
    const int row  = blockIdx.x * 16 + r;
    const float* L = logits + (size_t)row * CC;

    // split the 81 columns between the two lanes that share a row
    const int kbeg = half ? 41 : 0;
    const int kend = half ? 81 : 41;

    // row max (pairwise combine across the two half-lanes of this row)
    float m = -INFINITY;
    for (int k = kbeg; k < kend; ++k) m = fmaxf(m, L[k]);
    m = fmaxf(m, __shfl_xor(m, 16, 32));

    // exp tile into LDS; pad columns 81..83 with zeros
    for (int k = kbeg; k < kend; ++k) s_exp[r][k] = expf(L[k] - m);
    if (half) { s_exp[r][81] = 0.0f; s_exp[r][82] = 0.0f; s_exp[r][83] = 0.0f; }
    __syncthreads();

#if __has_builtin(__builtin_amdgcn_wmma_f32_16x16x4_f32)
    v8f acc = {};
    v2f bones; bones.x = 1.0f; bones.y = 1.0f;     // all-ones B: layout-proof
    for (int kc = 0; kc < 21; ++kc) {
        const int k0 = kc * 4 + half * 2;
        v2f a; a.x = s_exp[r][k0]; a.y = s_exp[r][k0 + 1];
        acc = __builtin_amdgcn_wmma_f32_16x16x4_f32(
            /*neg_a=*/false, a, /*neg_b=*/false, bones,
            /*c_mod=*/(short)0, acc, /*reuse_a=*/false, /*reuse_b=*/false);
    }
    // D[m][n] == rowsum[m] for every n. C/D layout: VGPR j holds M=j for
    // lanes 0-15 and M=8+j for lanes 16-31.
    if (lane == 0)  { for (int j = 0; j < 8; ++j) s_sum[j]     = acc[j]; }
    if (lane == 16) { for (int j = 0; j < 8; ++j) s_sum[8 + j] = acc[j]; }
#else
    if (half == 0) {
        float s = 0.0f;
        for (int k = 0; k < 81; ++k) s += s_exp[r][k];
        s_sum[r] = s;
    }
#endif
    __syncthreads();

    const float denom = s_sum[r];
    for (int k = kbeg; k < kend; ++k)
        probT[(size_t)k * NN + row] = s_exp[r][k] / denom;
}

// ---------------------------------------------------------------------------
// Kernel 2: one workgroup (1024 threads, 32 waves) per class.
// Wave 0 kicks a TDM bulk transfer of all proposals (16 KB) into LDS; the
// DMA overlaps the bitonic sort. Decode then gathers proposals from LDS in
// sorted order, greedy NMS runs over the V threshold survivors.
// ---------------------------------------------------------------------------
__global__ __launch_bounds__(1024) void nms_per_class(const float* __restrict__ probT,
                                                      const float* __restrict__ box_reg,
                                                      const float* __restrict__ proposals,
                                                      float* __restrict__ scores_sorted,
                                                      float* __restrict__ boxes_sorted) {
    __shared__ float s_prop[NN * 4];   // TDM-staged proposals (16 KB)
    __shared__ float s_key[NN];
    __shared__ int   s_idx[NN];
    __shared__ float s_box[NN * 4];
    __shared__ float s_area[NN];
    __shared__ int   s_supp[NN];
    __shared__ int   s_cnt;

    const int c = blockIdx.x;     // class id 0..80
    const int t = threadIdx.x;    // 0..1023

#if HAVE_TDM
    if (t < 32) {                 // wave-uniform: only wave 0 issues the DMA
        const unsigned long long ga = (unsigned long long)(uintptr_t)proposals;
        const unsigned lds_addr = (unsigned)(size_t)(void*)s_prop;

        v4u g0;
        g0.x = 1u;                                   // count=1, user descriptor
        g0.y = lds_addr;                             // LDS dest (bytes)
        g0.z = (unsigned)(ga & 0xFFFFFFFFu);         // global_addr[31:0]
        g0.w = (unsigned)((ga >> 32) & 0x01FFFFFFu)  // global_addr[56:32]
             | (2u << 30);                           // type=2 ("image")

        v8i g1;
        g1[0] = (int)(2u << 16);                     // data_size=2 (4B), mask=0
        g1[1] = (int)((4096u & 0xFFFFu) << 16);      // tensor_dim0 lo16
        g1[2] = (int)((4096u >> 16) | (1u << 16));   // tensor_dim0 hi16, dim1=1
        g1[3] = (int)(4096u << 16);                  // dim1 hi=0, tile_dim0=4096
        g1[4] = 1;                                   // tile_dim1=1, tile_dim2=0
        g1[5] = 4096;                                // tensor_dim0_stride lo32
        g1[6] = 0;                                   // stride0 hi, stride1 lo
        g1[7] = 0;                                   // stride1 hi

        v4i z4 = {0, 0, 0, 0};                       // groups 2/3: 2-D tensor
#if __clang_major__ >= 23
        v8i z8 = {0, 0, 0, 0, 0, 0, 0, 0};
        __builtin_amdgcn_tensor_load_to_lds(g0, g1, z4, z4, z8, 0);
#else
        __builtin_amdgcn_tensor_load_to_lds(g0, g1, z4, z4, 0);
#endif
    }
#else
    // fallback: plain staged copy (visibility via the sort barriers below)
    {
        s_prop[t * 4 + 0] = proposals[t * 4 + 0];
        s_prop[t * 4 + 1] = proposals[t * 4 + 1];
        s_prop[t * 4 + 2] = proposals[t * 4 + 2];
        s_prop[t * 4 + 3] = proposals[t * 4 + 3];
    }
#endif

    // warm the regression rows we will need after the sort (global_prefetch_b8)
    __builtin_prefetch(box_reg + (size_t)t * 4 * CC + 4 * c, 0, 1);

    const float p     = probT[(size_t)c * NN + t];   // coalesced gather
    const bool  valid = (c != 0) && (p > SCORE_TH);
    s_key[t] = valid ? p : -INFINITY;
    s_idx[t] = t;
    if (t == 0) s_cnt = 0;
    __syncthreads();

    // bitonic sort, descending by key (TDM runs underneath this phase)
    for (unsigned k = 2; k <= NN; k <<= 1) {
        for (unsigned j = k >> 1; j > 0; j >>= 1) {
            const unsigned ixj = t ^ j;
            if (ixj > (unsigned)t) {
                const bool  up = ((t & k) == 0);
                const float a = s_key[t], b = s_key[ixj];
                const bool  do_swap = up ? (a < b) : (a > b);
                if (do_swap) {
                    s_key[t] = b; s_key[ixj] = a;
                    const int ia = s_idx[t];
                    s_idx[t] = s_idx[ixj]; s_idx[ixj] = ia;
                }
            }
            __syncthreads();
        }
    }

    // number of threshold survivors (all finite keys sort to the front)
    if (s_key[t] > 0.0f) atomicAdd(&s_cnt, 1);

#if HAVE_TDM
    if (t < 32) __builtin_amdgcn_s_wait_tensorcnt((short)0);  // data landed
#endif
    __syncthreads();
    const int V = s_cnt;

    // decode + clip box for sorted slot t (original proposal s_idx[t])
    {
        const int i = s_idx[t];
        const float x1 = s_prop[i * 4 + 0], y1 = s_prop[i * 4 + 1];
        const float x2 = s_prop[i * 4 + 2], y2 = s_prop[i * 4 + 3];
        const float w  = x2 - x1 + 1.0f;
        const float h  = y2 - y1 + 1.0f;
        const float cx = x1 + 0.5f * w;
        const float cy = y1 + 0.5f * h;

        const float* rp = box_reg + (size_t)i * 4 * CC + 4 * c;
        const float dx = rp[0] * 0.1f;
        const float dy = rp[1] * 0.1f;
        const float dw = fminf(rp[2] * 0.2f, XFORM_CLIP);
        const float dh = fminf(rp[3] * 0.2f, XFORM_CLIP);

        const float pcx = dx * w + cx;
        const float pcy = dy * h + cy;
        const float pw  = expf(dw) * w;
        const float ph  = expf(dh) * h;

        float bx1 = pcx - 0.5f * pw;
        float by1 = pcy - 0.5f * ph;
        float bx2 = pcx + 0.5f * pw - 1.0f;
        float by2 = pcy + 0.5f * ph - 1.0f;
        bx1 = fminf(fmaxf(bx1, 0.0f), WIMG - 1.0f);
        by1 = fminf(fmaxf(by1, 0.0f), HIMG - 1.0f);
        bx2 = fminf(fmaxf(bx2, 0.0f), WIMG - 1.0f);
        by2 = fminf(fmaxf(by2, 0.0f), HIMG - 1.0f);

        s_box[t * 4 + 0] = bx1; s_box[t * 4 + 1] = by1;
        s_box[t * 4 + 2] = bx2; s_box[t * 4 + 3] = by2;
        s_area[t] = (bx2 - bx1 + 1.0f) * (by2 - by1 + 1.0f);
        s_supp[t] = 0;

        float* ob = boxes_sorted + ((size_t)c * NN + t) * 4;
        ob[0] = bx1; ob[1] = by1; ob[2] = bx2; ob[3] = by2;
    }

    // greedy NMS over the V sorted survivors
    for (int i = 0; i < V; ++i) {
        __syncthreads();
        if (s_supp[i] == 0) {                 // uniform: same LDS word for all
            if (t > i && t < V && s_supp[t] == 0) {
                const float ix1 = s_box[i * 4 + 0], iy1 = s_box[i * 4 + 1];
                const float ix2 = s_box[i * 4 + 2], iy2 = s_box[i * 4 + 3];
                const float lt_x = fmaxf(ix1, s_box[t * 4 + 0]);
                const float lt_y = fmaxf(iy1, s_box[t * 4 + 1]);
                const float rb_x = fminf(ix2, s_box[t * 4 + 2]);
                const float rb_y = fminf(iy2, s_box[t * 4 + 3]);
                const float iw = fmaxf(rb_x - lt_x + 1.0f, 0.0f);
                const float ih = fmaxf(rb_y - lt_y + 1.0f, 0.0f);
                const float inter = iw * ih;
                const float iou = inter / (s_area[i] + s_area[t] - inter);
                if (iou > NMS_TH) s_supp[t] = 1;
            }
        }
    }
    __syncthreads();

    scores_sorted[(size_t)c * NN + t] =
        (t < V && s_supp[t] == 0) ? s_key[t] : -INFINITY;
}

// ---------------------------------------------------------------------------
// Kernel 3: global top-100 over the 81*1024 flat kept-scores.
// 100 selection rounds; (score desc, index asc) comparator matches
// lax.top_k tie-breaking. Non-finite winners produce zero rows, like the
// reference's isfinite masking. scores_sorted is fully rewritten by K2 every
// call, so mutating it here is replay-safe.
// ---------------------------------------------------------------------------
__global__ __launch_bounds__(1024) void topk_gather(const float* __restrict__ boxes_sorted,
                                                    float* __restrict__ scores,
                                                    float* __restrict__ out) {
    __shared__ float rk[NN];
    __shared__ int   ri[NN];
    const int t = threadIdx.x;
    const int M = CC * NN;

    for (int round = 0; round < DETS; ++round) {
        float bk = -INFINITY;
        int   bi = 0x7FFFFFFF;
        for (int e = t; e < M; e += NN) {
            const float v = scores[e];
            if (v > bk || (v == bk && e < bi)) { bk = v; bi = e; }
        }
        rk[t] = bk; ri[t] = bi;
        for (int s = NN / 2; s > 0; s >>= 1) {
            __syncthreads();
            if (t < s) {
                const float ok = rk[t + s];
                const int   oi = ri[t + s];
                if (ok > rk[t] || (ok == rk[t] && oi < ri[t])) {
                    rk[t] = ok; ri[t] = oi;
                }
            }
        }
        __syncthreads();
        if (t == 0) {
            const float sc  = rk[0];
            const int   idx = ri[0];
            float* o = out + (size_t)round * 6;
            if (isfinite(sc)) {
                const float* b = boxes_sorted + (size_t)idx * 4;
                o[0] = b[0]; o[1] = b[1]; o[2] = b[2]; o[3] = b[3];
                o[4] = sc;
                o[5] = (float)(idx >> 10);   // class = idx / 1024
            } else {
                o[0] = 0.0f; o[1] = 0.0f; o[2] = 0.0f;
                o[3] = 0.0f; o[4] = 0.0f; o[5] = 0.0f;
            }
            scores[idx] = -INFINITY;         // remove winner for next round
        }
        __syncthreads();
    }
}

// ---------------------------------------------------------------------------
extern "C" void kernel_launch(void* const* d_in, const int* in_sizes, int n_in,
                              void* d_out, int out_size, void* d_ws, size_t ws_size,
                              hipStream_t stream) {
    const float* class_logits   = (const float*)d_in[0];   // [1024, 81]
    const float* box_regression = (const float*)d_in[1];   // [1024, 324]
    const float* proposals      = (const float*)d_in[2];   // [1024, 4]
    float*       out            = (float*)d_out;           // [100, 6]

    // workspace layout (~2 MB), fully rewritten each call
    float* probT         = (float*)d_ws;                   // CC*NN (transposed)
    float* scores_sorted = probT + (size_t)CC * NN;        // CC*NN
    float* boxes_sorted  = scores_sorted + (size_t)CC * NN;// CC*NN*4

    softmax_rows <<<NN / 16, 32,   0, stream>>>(class_logits, probT);
    nms_per_class<<<CC,      NN,   0, stream>>>(probT, box_regression, proposals,
                                                scores_sorted, boxes_sorted);
    topk_gather  <<<1,       NN,   0, stream>>>(boxes_sorted, scores_sorted, out);
    (void)in_sizes; (void)n_in; (void)out_size; (void)ws_size;
}